// TinyNet_65360812311080
// MI455X (gfx1250) — compile-verified
//
#include <hip/hip_runtime.h>
#include <cstddef>

// Grid2Points gather + ReLU for MI455X (gfx1250).
// Memory-bound (~103 MB traffic, ~12.8 MFLOP): optimize the data-movement path.
//   Primary path : GLOBAL_LOAD_ASYNC_TO_LDS_B128 double-buffered pipeline
//                  (ASYNCcnt / s_wait_asynccnt), LDS -> ReLU -> NT b128 stores.
//   Fallback     : direct NT b128 gather if the async
//                  builtins are not available in this toolchain.

typedef __attribute__((ext_vector_type(4))) float f32x4;
typedef __attribute__((ext_vector_type(4))) int   i32x4;

#define CH            64   // channels per row
#define LANES_PER_PT  16   // 16 lanes x float4 = one 256B row
#define PTS_PER_GRP   4    // points per 16-lane group per pipeline stage
#define GRPS          16   // 256 threads / 16 lanes

#if defined(__HIP_DEVICE_COMPILE__) && \
    __has_builtin(__builtin_amdgcn_global_load_async_to_lds_b128) && \
    __has_builtin(__builtin_amdgcn_s_wait_asynccnt)
#define USE_ASYNC_LDS 1
// Builtin signature (from hipcc diagnostic): (v4i as1*, v4i as3*, Ii, Ii)
typedef __attribute__((address_space(1))) i32x4 gbl_i32x4;
typedef __attribute__((address_space(3))) i32x4 lds_i32x4;
#endif

__global__ __launch_bounds__(256) void gather_relu_kernel(
    const int*   __restrict__ coors,        // [n_points]
    const float* __restrict__ grid,         // [features_num, CH]
    const int*   __restrict__ points_num_p, // device scalar
    float*       __restrict__ out,          // [n_points, CH]
    int n_points, int features_num)
{
  const int points_num = *points_num_p;
  const int lane       = threadIdx.x & (LANES_PER_PT - 1);
  const int grp        = threadIdx.x / LANES_PER_PT;
  const int grp_id     = blockIdx.x * GRPS + grp;
  const int stride     = gridDim.x * GRPS * PTS_PER_GRP;
  const int base0      = grp_id * PTS_PER_GRP;

  // Load 4 indices for batch starting at b; -1 marks "emit zeros".
  auto load_rows = [&](int b) -> i32x4 {
    i32x4 r;
    if (b + PTS_PER_GRP <= n_points && b + PTS_PER_GRP <= points_num) {
      r = *(const i32x4*)(coors + b);
    } else {
#pragma unroll
      for (int i = 0; i < PTS_PER_GRP; ++i) {
        const int p = b + i;
        r[i] = (p < n_points && p < points_num) ? coors[p] : -1;
      }
    }
    return r;
  };

#if defined(USE_ASYNC_LDS)
  // Double-buffered async gather staging: 2 * 16 grps * 4 pts * 256B = 32 KB.
  __shared__ __align__(16) float lbuf[2][GRPS][PTS_PER_GRP][CH];

  // Issue one pipeline stage: 4 async b128 ops per wave (2 rows each).
  auto issue = [&](const i32x4& rows, int buf) {
#pragma unroll
    for (int i = 0; i < PTS_PER_GRP; ++i) {
      unsigned r = (unsigned)rows[i];
      if (r >= (unsigned)features_num) r = 0u;  // clamp; masked at consume
      const float* gsrc = grid + (size_t)r * CH + (size_t)lane * 4;
      float*       ldst = &lbuf[buf][grp][i][lane * 4];
      __builtin_amdgcn_global_load_async_to_lds_b128(
          (gbl_i32x4*)(void*)gsrc, (lds_i32x4*)(void*)ldst, 0, 0);
    }
  };

  i32x4 rows_cur = {};
  if (base0 < n_points) {
    rows_cur = load_rows(base0);
    issue(rows_cur, 0);
  }

  int buf = 0;
  for (int base = base0; base < n_points; base += stride) {
    const int nbase = base + stride;
    const bool have_nxt = nbase < n_points;

    i32x4 rows_nxt = {};
    if (have_nxt) {
      rows_nxt = load_rows(nbase);
      issue(rows_nxt, buf ^ 1);
      // warm L2 for the stage after next (sequential index stream)
      if (nbase + stride < n_points)
        __builtin_prefetch(coors + nbase + stride, 0, 1);
    }

    // Previous batch (4 ops) complete; next batch may remain in flight.
    if (have_nxt) __builtin_amdgcn_s_wait_asynccnt(4);
    else          __builtin_amdgcn_s_wait_asynccnt(0);
    asm volatile("" ::: "memory");  // keep LDS reads below the wait

#pragma unroll
    for (int i = 0; i < PTS_PER_GRP; ++i) {
      const int p = base + i;
      if (p < n_points) {
        const unsigned r = (unsigned)rows_cur[i];
        f32x4 v = {0.f, 0.f, 0.f, 0.f};
        if (r < (unsigned)features_num) {
          v = *(const f32x4*)&lbuf[buf][grp][i][lane * 4];
          v.x = __builtin_fmaxf(v.x, 0.f);
          v.y = __builtin_fmaxf(v.y, 0.f);
          v.z = __builtin_fmaxf(v.z, 0.f);
          v.w = __builtin_fmaxf(v.w, 0.f);
        }
        __builtin_nontemporal_store(v, (f32x4*)(out + (size_t)p * CH) + lane);
      }
    }
    asm volatile("" ::: "memory");  // reads done before buffer is reissued

    rows_cur = rows_nxt;
    buf ^= 1;
  }

#else  // ---------------- fallback: direct NT b128 gather ----------------

  for (int base = base0; base < n_points; base += stride) {
    i32x4 rows = load_rows(base);

    const int nbase = base + stride;
    if (nbase + PTS_PER_GRP <= n_points && nbase + PTS_PER_GRP <= points_num) {
      i32x4 nrows = *(const i32x4*)(coors + nbase);
#pragma unroll
      for (int i = 0; i < PTS_PER_GRP; ++i) {
        const unsigned r = (unsigned)nrows[i];
        if (r < (unsigned)features_num)
          __builtin_prefetch(grid + (size_t)r * CH + (size_t)lane * 4, 0, 1);
      }
    }

    f32x4 vals[PTS_PER_GRP];
#pragma unroll
    for (int i = 0; i < PTS_PER_GRP; ++i) {
      const unsigned r = (unsigned)rows[i];
      f32x4 v = {0.f, 0.f, 0.f, 0.f};
      if (r < (unsigned)features_num) {
        const f32x4* src = (const f32x4*)(grid + (size_t)r * CH);
        v = __builtin_nontemporal_load(src + lane);
        v.x = __builtin_fmaxf(v.x, 0.f);
        v.y = __builtin_fmaxf(v.y, 0.f);
        v.z = __builtin_fmaxf(v.z, 0.f);
        v.w = __builtin_fmaxf(v.w, 0.f);
      }
      vals[i] = v;
    }

#pragma unroll
    for (int i = 0; i < PTS_PER_GRP; ++i) {
      const int p = base + i;
      if (p < n_points)
        __builtin_nontemporal_store(vals[i], (f32x4*)(out + (size_t)p * CH) + lane);
    }
  }
#endif
}

extern "C" void kernel_launch(void* const* d_in, const int* in_sizes, int n_in,
                              void* d_out, int out_size, void* d_ws, size_t ws_size,
                              hipStream_t stream) {
  (void)d_ws; (void)ws_size; (void)n_in; (void)out_size;

  const int*   coors = (const int*)d_in[0];   // rv_coors  [n_points]
  const float* grid  = (const float*)d_in[1]; // rv_x_i    [features, 64]
  const int*   pnum  = (const int*)d_in[2];   // points_num (device scalar)
  float*       out   = (float*)d_out;

  const int n_points = in_sizes[0];
  const int features = in_sizes[1] / CH;

  const int block       = 256;
  const int pts_per_blk = GRPS * PTS_PER_GRP;  // 64
  int blocks = (n_points + pts_per_blk - 1) / pts_per_blk;
  if (blocks > 1024) blocks = 1024;  // grid-stride: ~3 pipeline stages/group
  if (blocks < 1) blocks = 1;

  gather_relu_kernel<<<blocks, block, 0, stream>>>(coors, grid, pnum, out,
                                                   n_points, features);
}